// GeoGAT_24781961298580
// MI455X (gfx1250) — compile-verified
//
#include <hip/hip_runtime.h>

#define NEG_SLOPE 0.2f

typedef float v2f __attribute__((ext_vector_type(2)));
typedef float v8f __attribute__((ext_vector_type(8)));

__device__ __forceinline__ float leaky(float a) {
    return a > 0.f ? a : NEG_SLOPE * a;
}

// float atomic max via monotone int/uint bit ordering (init must be -inf)
__device__ __forceinline__ void atomicMaxF(float* addr, float v) {
    if (v >= 0.f) atomicMax((int*)addr, __float_as_int(v));
    else          atomicMin((unsigned int*)addr, __float_as_uint(v));
}

__global__ void k_fill(float* __restrict__ p, int n, float v) {
    int i = blockIdx.x * blockDim.x + threadIdx.x;
    if (i < n) p[i] = v;
}

// h[N,16] = x[N,128] @ W[128,16] via V_WMMA_F32_16X16X4_F32 (exact fp32).
// One wave computes one 16(M)x16(N) tile; K-loop of 32 WMMA ops.
__global__ void k_gemm1(const float* __restrict__ x, const float* __restrict__ W,
                        float* __restrict__ h, int n_nodes, int num_tiles) {
    __shared__ float sW[128 * 16];
    for (int i = threadIdx.x; i < 128 * 16; i += blockDim.x) sW[i] = W[i];
    __syncthreads();

    const int lane = threadIdx.x & 31;
    const int wave = threadIdx.x >> 5;
    const int tile = blockIdx.x * (blockDim.x >> 5) + wave;
    if (tile >= num_tiles) return;   // wave-uniform: EXEC stays all-ones

    // A layout (16x4 f32): lanes 0-15 -> row = lane, K = {k, k+1};
    //                      lanes 16-31 -> row = lane-16, K = {k+2, k+3}
    const int row  = tile * 16 + (lane & 15);
    const int rrow = row < n_nodes ? row : n_nodes - 1;   // clamp (tail safety)
    const int koff = (lane >> 4) << 1;                    // 0 or 2
    const int col  = lane & 15;                           // B/D column

    const float* __restrict__ xrow = x + (size_t)rrow * 128;
    v8f acc = {};
#pragma unroll
    for (int k = 0; k < 128; k += 4) {
        v2f a, b;
        a[0] = xrow[k + koff];
        a[1] = xrow[k + koff + 1];
        b[0] = sW[(k + koff) * 16 + col];
        b[1] = sW[(k + koff + 1) * 16 + col];
        acc = __builtin_amdgcn_wmma_f32_16x16x4_f32(
            /*neg_a=*/false, a, /*neg_b=*/false, b,
            /*c_mod=*/(short)0, acc, /*reuse_a=*/false, /*reuse_b=*/false);
    }

    // D layout: VGPR r -> lanes 0-15: M=r, lanes 16-31: M=r+8; N = lane&15
    const int mrow = tile * 16 + (lane >> 4) * 8;
#pragma unroll
    for (int r = 0; r < 8; ++r) {
        int m = mrow + r;
        if (m < n_nodes) h[(size_t)m * 16 + col] = acc[r];
    }
}

// per-(node,head) attention coefficients: as[n,h] = <h[n,h,:], a_src[h,:]>
__global__ void k_alpha(const float* __restrict__ h, const float* __restrict__ a_s,
                        const float* __restrict__ a_d, float* __restrict__ as,
                        float* __restrict__ ad, int n_nodes, int H, int C) {
    int i = blockIdx.x * blockDim.x + threadIdx.x;
    if (i >= n_nodes * H) return;
    int n = i / H, hh = i % H;
    const float* hp = h + (size_t)n * H * C + hh * C;
    float s = 0.f, d = 0.f;
    for (int c = 0; c < C; ++c) {
        float v = hp[c];
        s += v * a_s[hh * C + c];
        d += v * a_d[hh * C + c];
    }
    as[i] = s;
    ad[i] = d;
}

__global__ void k_edge_max(const int* __restrict__ src, const int* __restrict__ dst,
                           const float* __restrict__ as, const float* __restrict__ ad,
                           float* __restrict__ segmax, int E0, int Etot, int H) {
    int e = blockIdx.x * blockDim.x + threadIdx.x;
    if (e >= Etot) return;
    int s = (e < E0) ? src[e] : (e - E0);
    int d = (e < E0) ? dst[e] : (e - E0);
    for (int hh = 0; hh < H; ++hh) {
        float a = leaky(as[s * H + hh] + ad[d * H + hh]);
        atomicMaxF(&segmax[d * H + hh], a);
    }
}

__global__ void k_edge_sum(const int* __restrict__ src, const int* __restrict__ dst,
                           const float* __restrict__ as, const float* __restrict__ ad,
                           const float* __restrict__ segmax, float* __restrict__ denom,
                           int E0, int Etot, int H) {
    int e = blockIdx.x * blockDim.x + threadIdx.x;
    if (e >= Etot) return;
    int s = (e < E0) ? src[e] : (e - E0);
    int d = (e < E0) ? dst[e] : (e - E0);
    for (int hh = 0; hh < H; ++hh) {
        float a = leaky(as[s * H + hh] + ad[d * H + hh]);
        atomicAdd(&denom[d * H + hh], __expf(a - segmax[d * H + hh]));
    }
}

__global__ void k_edge_aggr(const int* __restrict__ src, const int* __restrict__ dst,
                            const float* __restrict__ as, const float* __restrict__ ad,
                            const float* __restrict__ segmax, const float* __restrict__ denom,
                            const float* __restrict__ h, float* __restrict__ agg,
                            int E0, int Etot, int H, int C) {
    int e = blockIdx.x * blockDim.x + threadIdx.x;
    if (e >= Etot) return;
    int s = (e < E0) ? src[e] : (e - E0);
    int d = (e < E0) ? dst[e] : (e - E0);
    for (int hh = 0; hh < H; ++hh) {
        float a = leaky(as[s * H + hh] + ad[d * H + hh]);
        float p = __expf(a - segmax[d * H + hh]) / (denom[d * H + hh] + 1e-16f);
        const float* hp = h + (size_t)s * H * C + hh * C;
        float* op = agg + (size_t)d * H * C + hh * C;
        for (int c = 0; c < C; ++c) atomicAdd(&op[c], hp[c] * p);
    }
}

// layer-1 epilogue fused with layer-2 feature transform (K=16, OUT_CH=1):
// v = relu(agg1 + b1); z = v . W2; alpha2 = z * a_{src,dst}2
__global__ void k_post1(const float* __restrict__ agg1, const float* __restrict__ b1,
                        const float* __restrict__ W2, const float* __restrict__ a_src2,
                        const float* __restrict__ a_dst2, float* __restrict__ z,
                        float* __restrict__ as2, float* __restrict__ ad2, int n_nodes) {
    int n = blockIdx.x * blockDim.x + threadIdx.x;
    if (n >= n_nodes) return;
    float zz = 0.f;
#pragma unroll
    for (int j = 0; j < 16; ++j) {
        float v = agg1[(size_t)n * 16 + j] + b1[j];
        v = fmaxf(v, 0.f);
        zz += v * W2[j];
    }
    z[n] = zz;
    as2[n] = zz * a_src2[0];
    ad2[n] = zz * a_dst2[0];
}

__global__ void k_final(const float* __restrict__ agg2, const float* __restrict__ b2,
                        float* __restrict__ out, int n_nodes) {
    int n = blockIdx.x * blockDim.x + threadIdx.x;
    if (n >= n_nodes) return;
    out[n] = agg2[n] + b2[0];   // mean over 1 head == identity
}

extern "C" void kernel_launch(void* const* d_in, const int* in_sizes, int n_in,
                              void* d_out, int out_size, void* d_ws, size_t ws_size,
                              hipStream_t stream) {
    const float* x      = (const float*)d_in[0];
    const int*   ei     = (const int*)d_in[1];
    const float* W1     = (const float*)d_in[2];
    const float* a_src1 = (const float*)d_in[3];
    const float* a_dst1 = (const float*)d_in[4];
    const float* b1     = (const float*)d_in[5];
    const float* W2     = (const float*)d_in[6];
    const float* a_src2 = (const float*)d_in[7];
    const float* a_dst2 = (const float*)d_in[8];
    const float* b2     = (const float*)d_in[9];

    const int Nn   = in_sizes[0] / 128;   // 100000 nodes
    const int E0   = in_sizes[1] / 2;     // 6400000 edges
    const int Etot = E0 + Nn;             // + self loops
    const int* src = ei;
    const int* dst = ei + E0;

    float* ws = (float*)d_ws;
    float* h1    = ws; ws += (size_t)Nn * 16;
    float* as1   = ws; ws += (size_t)Nn * 2;
    float* ad1   = ws; ws += (size_t)Nn * 2;
    float* smax1 = ws; ws += (size_t)Nn * 2;
    float* den1  = ws; ws += (size_t)Nn * 2;
    float* agg1  = ws; ws += (size_t)Nn * 16;
    float* z     = ws; ws += Nn;
    float* as2   = ws; ws += Nn;
    float* ad2   = ws; ws += Nn;
    float* smax2 = ws; ws += Nn;
    float* den2  = ws; ws += Nn;
    float* agg2  = ws; ws += Nn;

    const float NEG_INF = -__builtin_inff();
    auto cdiv = [](int a, int b) { return (a + b - 1) / b; };

    // ---- init accumulators (every call: deterministic) ----
    k_fill<<<cdiv(Nn * 2, 256), 256, 0, stream>>>(smax1, Nn * 2, NEG_INF);
    k_fill<<<cdiv(Nn * 2, 256), 256, 0, stream>>>(den1, Nn * 2, 0.f);
    k_fill<<<cdiv(Nn * 16, 256), 256, 0, stream>>>(agg1, Nn * 16, 0.f);
    k_fill<<<cdiv(Nn, 256), 256, 0, stream>>>(smax2, Nn, NEG_INF);
    k_fill<<<cdiv(Nn, 256), 256, 0, stream>>>(den2, Nn, 0.f);
    k_fill<<<cdiv(Nn, 256), 256, 0, stream>>>(agg2, Nn, 0.f);

    // ---- layer 1 ----
    const int tiles = cdiv(Nn, 16);
    const int wavesPerBlk = 2;            // 64 threads/block, 2 tiles/block
    k_gemm1<<<cdiv(tiles, wavesPerBlk), 32 * wavesPerBlk, 0, stream>>>(x, W1, h1, Nn, tiles);

    k_alpha<<<cdiv(Nn * 2, 256), 256, 0, stream>>>(h1, a_src1, a_dst1, as1, ad1, Nn, 2, 8);

    const int eg = cdiv(Etot, 256);
    k_edge_max <<<eg, 256, 0, stream>>>(src, dst, as1, ad1, smax1, E0, Etot, 2);
    k_edge_sum <<<eg, 256, 0, stream>>>(src, dst, as1, ad1, smax1, den1, E0, Etot, 2);
    k_edge_aggr<<<eg, 256, 0, stream>>>(src, dst, as1, ad1, smax1, den1, h1, agg1,
                                        E0, Etot, 2, 8);

    // ---- layer-1 epilogue + layer-2 transform ----
    k_post1<<<cdiv(Nn, 256), 256, 0, stream>>>(agg1, b1, W2, a_src2, a_dst2,
                                               z, as2, ad2, Nn);

    // ---- layer 2 (H=1, C=1, feature == z) ----
    k_edge_max <<<eg, 256, 0, stream>>>(src, dst, as2, ad2, smax2, E0, Etot, 1);
    k_edge_sum <<<eg, 256, 0, stream>>>(src, dst, as2, ad2, smax2, den2, E0, Etot, 1);
    k_edge_aggr<<<eg, 256, 0, stream>>>(src, dst, as2, ad2, smax2, den2, z, agg2,
                                        E0, Etot, 1, 1);

    k_final<<<cdiv(Nn, 256), 256, 0, stream>>>(agg2, b2, (float*)d_out, Nn);
}